// Gate_7378753814906
// MI455X (gfx1250) — compile-verified
//
#include <hip/hip_runtime.h>
#include <math.h>

typedef __attribute__((ext_vector_type(2))) float v2f;
typedef __attribute__((ext_vector_type(8))) float v8f;
typedef __attribute__((ext_vector_type(2))) int   v2i;

#define T_TOKENS 32768
#define DDIM     4096
#define NEXP     8
#define WAVES_PER_BLOCK  8
#define TOKENS_PER_WAVE  16
#define TOKENS_PER_BLOCK (WAVES_PER_BLOCK * TOKENS_PER_WAVE)
#define KSTEPS (DDIM / 4)   // 1024 WMMA steps of K=4
#define GROUP  8            // K-steps per pipeline stage

__global__ __launch_bounds__(256) void gate_wmma_kernel(
    const float* __restrict__ x,      // [T, D]
    const float* __restrict__ w,      // [E, D]
    const float* __restrict__ bias,   // [E]
    float* __restrict__ outw,         // [T, 2]
    int*   __restrict__ outi)         // [T, 2]
{
    __shared__ float lds_tile[WAVES_PER_BLOCK][TOKENS_PER_WAVE][17];

    const int lane = threadIdx.x & 31;
    const int wave = threadIdx.x >> 5;
    const int n    = lane & 15;      // tile-local row (A) / column (B)
    const int half = lane >> 4;      // 0 or 1
    const int koff = half * 2;       // K offset within the 4-wide step

    const int tileBase = blockIdx.x * TOKENS_PER_BLOCK + wave * TOKENS_PER_WAVE;

    const float* xp = x + (size_t)(tileBase + n) * DDIM + koff;   // A stream (HBM)
    const float* wp = w + (size_t)(n & (NEXP - 1)) * DDIM + koff; // B stream (cache-hot)

    v8f acc0 = {};
    v8f acc1 = {};

    // Two register-resident fragment buffers -> 32 loads in flight per wave.
    v2f afA[GROUP], bfA[GROUP];
    v2f afB[GROUP], bfB[GROUP];

#define LOAD_GROUP(AF, BF, KB)                                     \
    {                                                              \
        _Pragma("unroll")                                          \
        for (int u = 0; u < GROUP; ++u) {                          \
            (AF)[u] = *(const v2f*)(xp + ((KB) + u) * 4);          \
            (BF)[u] = *(const v2f*)(wp + ((KB) + u) * 4);          \
        }                                                          \
    }

#define WMMA_GROUP(AF, BF)                                         \
    {                                                              \
        _Pragma("unroll")                                          \
        for (int u = 0; u < GROUP; ++u) {                          \
            if (u & 1)                                             \
                acc1 = __builtin_amdgcn_wmma_f32_16x16x4_f32(      \
                    false, (AF)[u], false, (BF)[u], (short)0,      \
                    acc1, false, false);                           \
            else                                                   \
                acc0 = __builtin_amdgcn_wmma_f32_16x16x4_f32(      \
                    false, (AF)[u], false, (BF)[u], (short)0,      \
                    acc0, false, false);                           \
        }                                                          \
    }

    LOAD_GROUP(afA, bfA, 0);
    LOAD_GROUP(afB, bfB, GROUP);

    // Steady state: kb = 0, 16, ..., 992. Consume groups kb, kb+8 while
    // loading groups kb+16, kb+24. Tail (1008, 1016) already resident.
    for (int kb = 0; kb + 2 * GROUP < KSTEPS; kb += 2 * GROUP) {
        __builtin_prefetch(xp + kb * 4 + 1024, 0, 1);  // ~4KB ahead per row
        WMMA_GROUP(afA, bfA);
        LOAD_GROUP(afA, bfA, kb + 2 * GROUP);
        WMMA_GROUP(afB, bfB);
        LOAD_GROUP(afB, bfB, kb + 3 * GROUP);
    }
    WMMA_GROUP(afA, bfA);
    WMMA_GROUP(afB, bfB);

#undef LOAD_GROUP
#undef WMMA_GROUP

    v8f acc = acc0 + acc1;

    // C layout: VGPR j, lanes 0-15 -> row j, lanes 16-31 -> row j+8; col = lane%16.
#pragma unroll
    for (int j = 0; j < 8; ++j) {
        lds_tile[wave][j + 8 * half][n] = acc[j];
    }
    // Wave-local LDS RAW fence (DS ops are per-wave in-order; explicit wait is cheap/safe).
    asm volatile("s_wait_dscnt 0x0" ::: "memory");

    if (lane < TOKENS_PER_WAVE) {
        const int t   = lane;
        const int tok = tileBase + t;

        float sc[NEXP];
        float bsc[NEXP];
#pragma unroll
        for (int e = 0; e < NEXP; ++e) {
            float z  = lds_tile[wave][t][e];
            // numerically-safe softplus; logits here are small (|z| ~ few)
            float sp = (z > 15.0f) ? z : log1pf(__expf(z));
            float s  = sqrtf(sp);
            sc[e]  = s;
            bsc[e] = s + bias[e];
        }

        // top-2 over biased scores; strict '>' => lowest index wins ties (matches lax.top_k)
        float b1 = -1e30f, w1 = 0.0f; int i1 = 0;
#pragma unroll
        for (int e = 0; e < NEXP; ++e) {
            if (bsc[e] > b1) { b1 = bsc[e]; w1 = sc[e]; i1 = e; }
        }
        float b2 = -1e30f, w2 = 0.0f; int i2 = 0;
#pragma unroll
        for (int e = 0; e < NEXP; ++e) {
            if (e != i1 && bsc[e] > b2) { b2 = bsc[e]; w2 = sc[e]; i2 = e; }
        }

        const float rs = 1.0f / (w1 + w2);   // ROUTE_SCALE == 1.0
        v2f ow; ow.x = w1 * rs; ow.y = w2 * rs;
        v2i oi; oi.x = i1;      oi.y = i2;
        *(v2f*)(outw + 2 * (size_t)tok) = ow;
        *(v2i*)(outi + 2 * (size_t)tok) = oi;
    }
}

extern "C" void kernel_launch(void* const* d_in, const int* in_sizes, int n_in,
                              void* d_out, int out_size, void* d_ws, size_t ws_size,
                              hipStream_t stream) {
    (void)in_sizes; (void)n_in; (void)out_size; (void)d_ws; (void)ws_size;

    const float* x    = (const float*)d_in[0];   // [32768, 4096] f32
    const float* wgt  = (const float*)d_in[1];   // [8, 4096] f32
    const float* bias = (const float*)d_in[2];   // [8] f32

    float* outw = (float*)d_out;                       // weights [T,2] f32
    int*   outi = (int*)d_out + (size_t)T_TOKENS * 2;  // indices [T,2] i32

    dim3 grid(T_TOKENS / TOKENS_PER_BLOCK);  // 256 blocks
    dim3 block(256);                         // 8 waves of 32
    gate_wmma_kernel<<<grid, block, 0, stream>>>(x, wgt, bias, outw, outi);
}